// CBMF_446676598939
// MI455X (gfx1250) — compile-verified
//
#include <hip/hip_runtime.h>

typedef float v2f __attribute__((ext_vector_type(2)));
typedef float v8f __attribute__((ext_vector_type(8)));

// Fused: gather embed rows -> per-row dot via V_WMMA_F32_16X16X4_F32 diagonal
// -> write predict, and materialize u / it rows from the same registers.
//
// One wave32 handles a 16-row tile:
//   A (16x4 f32): lane L<16 -> M=L, K-pair {k0,k0+1}; lane>=16 -> M=L-16, {k0+2,k0+3}
//   B (4x16 f32): mirrored layout -> identical float2 addressing on the It tile
//   D (16x16 f32): diag row v in acc[v] at lane v (M<8) / lane v+24 (M>=8)
__global__ __launch_bounds__(256) void cbmf_fused_wmma(
    const int*   __restrict__ user,
    const int*   __restrict__ item,
    const float* __restrict__ average,
    const float* __restrict__ embed_user,
    const float* __restrict__ embed_item,
    const float* __restrict__ user_bias,
    const float* __restrict__ item_bias,
    float*       __restrict__ out_pred,
    float*       __restrict__ out_u,
    float*       __restrict__ out_it,
    int n_rows)
{
    const int lane = threadIdx.x & 31;
    const int wave = (blockIdx.x * blockDim.x + threadIdx.x) >> 5;
    const int row_base = wave * 16;
    if (row_base >= n_rows) return;          // uniform per wave: EXEC stays all-1s

    const int sub  = lane & 15;              // M for A-tile, N for B-tile
    const int koff = (lane >> 4) << 1;       // 0 or 2: which K pair this lane owns

    const int r    = row_base + sub;
    const int urow = user[r];
    const int irow = item[r];

    const float* uptr = embed_user + (size_t)urow * 128 + koff;
    const float* iptr = embed_item + (size_t)irow * 128 + koff;
    float* ou = out_u  + (size_t)r * 128 + koff;
    float* oi = out_it + (size_t)r * 128 + koff;

    v8f acc = {};
    #pragma unroll
    for (int k0 = 0; k0 < 128; k0 += 4) {
        v2f a = *(const v2f*)(uptr + k0);    // U[M][k0+koff .. +1]
        v2f b = *(const v2f*)(iptr + k0);    // It[N][k0+koff .. +1] == B^T slice
        *(v2f*)(ou + k0) = a;                // materialize u row (register reuse)
        *(v2f*)(oi + k0) = b;                // materialize it row
        // 8-arg form: (neg_a, A, neg_b, B, c_mod, C, reuse_a, reuse_b)
        acc = __builtin_amdgcn_wmma_f32_16x16x4_f32(
            false, a, false, b, (short)0, acc, false, false);
    }

    // Pull the diagonal: row v -> acc[v] at lane v (rows 0..7) / lane v+24 (rows 8..15).
    float dv = 0.0f;
    #pragma unroll
    for (int v = 0; v < 8; ++v) {
        if (lane == v || lane == v + 24) dv = acc[v];
    }
    if (lane < 8 || lane >= 24) {
        // this lane's own urow/irow correspond to its diag row (r == row_base + sub)
        float pred = dv * 0.7f + average[0] + user_bias[urow] + item_bias[irow];
        out_pred[r] = pred;
    }
}

extern "C" void kernel_launch(void* const* d_in, const int* in_sizes, int n_in,
                              void* d_out, int out_size, void* d_ws, size_t ws_size,
                              hipStream_t stream) {
    (void)n_in; (void)out_size; (void)d_ws; (void)ws_size;

    const int*   user       = (const int*)  d_in[0];
    const int*   item       = (const int*)  d_in[1];
    const float* average    = (const float*)d_in[2];
    const float* embed_user = (const float*)d_in[3];
    const float* embed_item = (const float*)d_in[4];
    const float* user_bias  = (const float*)d_in[5];
    const float* item_bias  = (const float*)d_in[6];

    const int B = in_sizes[0];               // 131072 (divisible by 16)
    float* out      = (float*)d_out;
    float* out_pred = out;                           // [B]
    float* out_u    = out + B;                       // [B,128]
    float* out_it   = out + B + (size_t)B * 128;     // [B,128]

    const int waves   = B / 16;              // one wave per 16-row tile
    const int threads = waves * 32;
    const int block   = 256;                 // 8 waves / block (wave32)
    const int grid    = (threads + block - 1) / block;

    cbmf_fused_wmma<<<grid, block, 0, stream>>>(
        user, item, average, embed_user, embed_item, user_bias, item_bias,
        out_pred, out_u, out_it, B);
}